// GaussianRenderer_64665027608917
// MI455X (gfx1250) — compile-verified
//
#include <hip/hip_runtime.h>
#include <math.h>

#define NGAUSS 1024
#define IMH 256
#define IMW 256
#define REC 12   // floats per record (48 B, 16B-aligned)
// record layout: [0]=mx [1]=my [2]=ia [3]=ib [4]=ic [5]=norm(=opac*valid/(2pi*sqrt(det))) [6..8]=rgb [9..11]=pad

// ---------------- LDS address helper (32-bit LDS offset for async ops) ----------------
typedef __attribute__((address_space(3))) float lds_f32;
static __device__ __forceinline__ unsigned lds_off(void* p) {
  return (unsigned)(unsigned long long)(lds_f32*)p;
}

// ---------------- Kernel 1: per-gaussian preprocessing ----------------
__global__ void __launch_bounds__(256) gs_preprocess(
    const float* __restrict__ means3D, const float* __restrict__ covs3d,
    const float* __restrict__ colors,  const float* __restrict__ opac,
    const float* __restrict__ K, const float* __restrict__ R, const float* __restrict__ t,
    float* __restrict__ rec_raw, float* __restrict__ depth_out, int n)
{
  int i = blockIdx.x * blockDim.x + threadIdx.x;
  if (i >= n) return;
  float m0 = means3D[3*i+0], m1 = means3D[3*i+1], m2 = means3D[3*i+2];
  // cam = R*m + t
  float cam0 = R[0]*m0 + R[1]*m1 + R[2]*m2 + t[0];
  float cam1 = R[3]*m0 + R[4]*m1 + R[5]*m2 + t[1];
  float cam2 = R[6]*m0 + R[7]*m1 + R[8]*m2 + t[2];
  float depth = fmaxf(cam2, 1.0f);
  // screen = K*cam
  float s0 = K[0]*cam0 + K[1]*cam1 + K[2]*cam2;
  float s1 = K[3]*cam0 + K[4]*cam1 + K[5]*cam2;
  float s2 = K[6]*cam0 + K[7]*cam1 + K[8]*cam2;
  float inv_s2 = 1.0f / s2;
  float mx = s0 * inv_s2, my = s1 * inv_s2;
  float fx = K[0], fy = K[4];
  float invz  = 1.0f / cam2;
  float invz2 = invz * invz;
  // J (2x3 projection Jacobian); J01 = J10 = 0
  float J00 = fx * invz, J02 = -cam0 * invz2 * fx;
  float J11 = fy * invz, J12 = -cam1 * invz2 * fy;
  // covcam = R * C * R^T
  const float* C = covs3d + 9*i;
  float RC[9], CC[9];
  #pragma unroll
  for (int r = 0; r < 3; ++r)
    #pragma unroll
    for (int c = 0; c < 3; ++c)
      RC[3*r+c] = R[3*r+0]*C[0+c] + R[3*r+1]*C[3+c] + R[3*r+2]*C[6+c];
  #pragma unroll
  for (int r = 0; r < 3; ++r)
    #pragma unroll
    for (int c = 0; c < 3; ++c)
      CC[3*r+c] = RC[3*r+0]*R[3*c+0] + RC[3*r+1]*R[3*c+1] + RC[3*r+2]*R[3*c+2];
  // M = J*CC (2x3); cov2 = M*J^T (2x2)
  float M00 = J00*CC[0] + J02*CC[6];
  float M01 = J00*CC[1] + J02*CC[7];
  float M02 = J00*CC[2] + J02*CC[8];
  float M10 = J11*CC[3] + J12*CC[6];
  float M11 = J11*CC[4] + J12*CC[7];
  float M12 = J11*CC[5] + J12*CC[8];
  float a = M00*J00 + M02*J02 + 1e-4f;
  float b = M01*J11 + M02*J12;
  float c = M11*J11 + M12*J12 + 1e-4f;
  float det = a*c - b*b;
  float inv_det = 1.0f / det;
  float ia =  c * inv_det;
  float ib = -b * inv_det;
  float ic =  a * inv_det;
  bool  valid = (depth > 1.0f) && (depth < 50.0f);
  float norm  = valid ? (opac[i] / (6.283185307179586f * sqrtf(det))) : 0.0f;
  float* rec = rec_raw + REC * i;
  rec[0] = mx;  rec[1] = my;  rec[2] = ia;  rec[3] = ib;
  rec[4] = ic;  rec[5] = norm;
  rec[6] = colors[3*i+0]; rec[7] = colors[3*i+1]; rec[8] = colors[3*i+2];
  rec[9] = 0.f; rec[10] = 0.f; rec[11] = 0.f;
  depth_out[i] = depth;
}

// ---------------- Kernel 2: bitonic argsort of depths (stable via index tie-break) ----------------
__global__ void __launch_bounds__(NGAUSS) gs_sort(
    const float* __restrict__ depth, int* __restrict__ order, int n)
{
  __shared__ float sk[NGAUSS];
  __shared__ int   sv[NGAUSS];
  int tid = threadIdx.x;
  sk[tid] = (tid < n) ? depth[tid] : 3.4e38f;
  sv[tid] = tid;
  __syncthreads();
  for (int size = 2; size <= NGAUSS; size <<= 1) {
    for (int stride = size >> 1; stride > 0; stride >>= 1) {
      int partner = tid ^ stride;
      if (partner > tid) {
        bool asc = (tid & size) == 0;
        float ka = sk[tid], kb = sk[partner];
        int   va = sv[tid], vb = sv[partner];
        bool agtb = (ka > kb) || ((ka == kb) && (va > vb));
        if (asc == agtb) {
          sk[tid] = kb; sk[partner] = ka;
          sv[tid] = vb; sv[partner] = va;
        }
      }
      __syncthreads();
    }
  }
  if (tid < n) order[tid] = sv[tid];
}

// ---------------- Kernel 3: gather records into sorted order ----------------
__global__ void __launch_bounds__(256) gs_gather(
    const float* __restrict__ rec_raw, const int* __restrict__ order,
    float* __restrict__ rec_sorted, int n)
{
  int i = blockIdx.x * blockDim.x + threadIdx.x;
  if (i >= n) return;
  int src = order[i];
  const float4* s = (const float4*)(rec_raw + REC * src);
  float4* d = (float4*)(rec_sorted + REC * i);
  d[0] = s[0]; d[1] = s[1]; d[2] = s[2];
}

// ---------------- Kernel 4: per-pixel front-to-back compositing ----------------
// 128 blocks x 256 threads; each thread composites TWO pixels (rows y and y+128)
// -> two independent transmittance chains for ILP, hides v_exp_f32 latency.
__global__ void __launch_bounds__(IMW) gs_render(
    const float* __restrict__ rec_sorted, float* __restrict__ out, int n)
{
  __shared__ __align__(16) float lds[NGAUSS * REC];  // 48 KB

  // Stage entire sorted splat table into LDS via CDNA5 async DMA (ASYNCcnt path).
  // 1024*12 floats = 3072 b128 chunks = 256 threads x 12.
  unsigned base = lds_off(&lds[0]);
  #pragma unroll
  for (int r = 0; r < 12; ++r) {
    unsigned byteoff = (unsigned)((r * IMW + threadIdx.x) * 16);
    unsigned la = base + byteoff;
    asm volatile("global_load_async_to_lds_b128 %0, %1, %2 offset:0"
                 :: "v"(la), "v"(byteoff), "s"(rec_sorted)
                 : "memory");
  }
  asm volatile("s_wait_asynccnt 0" ::: "memory");
  __syncthreads();

  float px  = (float)threadIdx.x;            // column
  float py0 = (float)blockIdx.x;             // row (top half)
  float py1 = (float)(blockIdx.x + IMH / 2); // row (bottom half)
  float T0 = 1.0f, T1 = 1.0f;
  float aR0 = 0.f, aG0 = 0.f, aB0 = 0.f;
  float aR1 = 0.f, aG1 = 0.f, aB1 = 0.f;

  #pragma unroll 4
  for (int g = 0; g < n; ++g) {
    int rb = g * REC;
    float4 r0 = *(const float4*)&lds[rb + 0];   // mx,my,ia,ib  (LDS broadcast)
    float4 r1 = *(const float4*)&lds[rb + 4];   // ic,norm,cr,cg
    float  cb = lds[rb + 8];

    float dx  = px  - r0.x;
    float dy0 = py0 - r0.y;
    float dy1 = py1 - r0.y;
    float qx  = r0.z * dx * dx;        // ia*dx^2 (shared between both pixels)
    float tb  = 2.0f * r0.w * dx;      // 2*ib*dx (shared)
    float q0  = fmaf(r1.x * dy0, dy0, fmaf(tb, dy0, qx));
    float q1  = fmaf(r1.x * dy1, dy1, fmaf(tb, dy1, qx));

    float al0 = r1.y * __expf(-0.5f * q0);
    float al1 = r1.y * __expf(-0.5f * q1);

    float w0 = T0 * al0;
    float w1 = T1 * al1;
    aR0 = fmaf(w0, r1.z, aR0);  aR1 = fmaf(w1, r1.z, aR1);
    aG0 = fmaf(w0, r1.w, aG0);  aG1 = fmaf(w1, r1.w, aG1);
    aB0 = fmaf(w0, cb,   aB0);  aB1 = fmaf(w1, cb,   aB1);
    T0  = fmaf(-T0, al0, T0);   // T *= (1 - alpha)
    T1  = fmaf(-T1, al1, T1);
  }

  int pix0 = (int)blockIdx.x * IMW + (int)threadIdx.x;
  int pix1 = ((int)blockIdx.x + IMH / 2) * IMW + (int)threadIdx.x;
  out[3*pix0 + 0] = aR0;  out[3*pix0 + 1] = aG0;  out[3*pix0 + 2] = aB0;
  out[3*pix1 + 0] = aR1;  out[3*pix1 + 1] = aG1;  out[3*pix1 + 2] = aB1;
}

// ---------------- Host launcher ----------------
extern "C" void kernel_launch(void* const* d_in, const int* in_sizes, int n_in,
                              void* d_out, int out_size, void* d_ws, size_t ws_size,
                              hipStream_t stream) {
  const float* means3D = (const float*)d_in[0];
  const float* covs3d  = (const float*)d_in[1];
  const float* colors  = (const float*)d_in[2];
  const float* opac    = (const float*)d_in[3];
  const float* K       = (const float*)d_in[4];
  const float* R       = (const float*)d_in[5];
  const float* t       = (const float*)d_in[6];
  int n = in_sizes[3];                // opacities: N elements
  if (n > NGAUSS) n = NGAUSS;

  float* ws         = (float*)d_ws;
  float* rec_raw    = ws;                         // NGAUSS*REC floats
  float* rec_sorted = ws + NGAUSS * REC;          // NGAUSS*REC floats
  float* depth      = ws + 2 * NGAUSS * REC;      // NGAUSS floats
  int*   order      = (int*)(ws + 2 * NGAUSS * REC + NGAUSS);  // NGAUSS ints

  gs_preprocess<<<(n + 255) / 256, 256, 0, stream>>>(
      means3D, covs3d, colors, opac, K, R, t, rec_raw, depth, n);
  gs_sort<<<1, NGAUSS, 0, stream>>>(depth, order, n);
  gs_gather<<<(n + 255) / 256, 256, 0, stream>>>(rec_raw, order, rec_sorted, n);
  gs_render<<<IMH / 2, IMW, 0, stream>>>(rec_sorted, (float*)d_out, n);
}